// LowerBodyScore_3229815406606
// MI455X (gfx1250) — compile-verified
//
#include <hip/hip_runtime.h>
#include <math.h>

// Soft-DTW on MI455X (gfx1250, wave32).
// Phase 1: prep/mask -> f1p,f2p [4096 x 72] fp32
// Phase 2: row norms
// Phase 3: D = n1 + n2^T - 2*f1@f2^T via V_WMMA_F32_16X16X4_F32 (full fp32, matches ref)
// Phase 4: anti-diagonal soft-DTW DP, single WGP, rolling diagonals in LDS,
//          software-pipelined L2 gather of the next diagonal.

typedef __attribute__((ext_vector_type(2))) float v2f;
typedef __attribute__((ext_vector_type(8))) float v8f;

#define T_LEN 4096
#define FDIM  72
#define BIGF  1e10f
// joints {3,6,9,12,13,14,15,16,17,18,19,20,21,22,23} zeroed
#define IGNORE_MASK 0x00FFF248u

// ---------------- Phase 1: mask + transpose to [T, J*C] ----------------
__global__ void sdtw_prep(const float* __restrict__ x,
                          const float* __restrict__ y,
                          float* __restrict__ f1p,
                          float* __restrict__ f2p) {
  int idx = blockIdx.x * blockDim.x + threadIdx.x;
  const int per = T_LEN * FDIM;
  if (idx >= 2 * per) return;
  const float* src = (idx < per) ? x : y;
  float* dst       = (idx < per) ? f1p : f2p;
  int rem = (idx < per) ? idx : idx - per;
  int jc = rem / T_LEN;           // consecutive threads walk t -> coalesced reads
  int t  = rem - jc * T_LEN;
  int joint = jc / 3;
  float v = ((IGNORE_MASK >> joint) & 1u) ? 0.0f : src[jc * T_LEN + t];
  dst[t * FDIM + jc] = v;         // [T, 72] row-major
}

// ---------------- Phase 2: row squared norms (one wave32 per row) ----------------
__global__ void sdtw_norms(const float* __restrict__ f1p,
                           const float* __restrict__ f2p,
                           float* __restrict__ n1,
                           float* __restrict__ n2) {
  int wave = (blockIdx.x * blockDim.x + threadIdx.x) >> 5;
  int lane = threadIdx.x & 31;
  if (wave >= 2 * T_LEN) return;
  const float* f = (wave < T_LEN) ? f1p : f2p;
  float* n       = (wave < T_LEN) ? n1  : n2;
  int row        = (wave < T_LEN) ? wave : wave - T_LEN;
  float s = 0.0f;
  for (int c = lane; c < FDIM; c += 32) {
    float v = f[row * FDIM + c];
    s += v * v;
  }
  for (int off = 16; off; off >>= 1) s += __shfl_xor(s, off, 32);
  if (lane == 0) n[row] = s;
}

// ---------------- Phase 3: D[i,j] = n1[i]+n2[j]-2*(f1 f2^T) via fp32 WMMA ----------------
// One wave per 16x16 output tile. K=72 -> 18 steps of V_WMMA_F32_16X16X4_F32.
__global__ void __launch_bounds__(256)
sdtw_dist_wmma(const float* __restrict__ f1p,
               const float* __restrict__ f2p,
               const float* __restrict__ n1,
               const float* __restrict__ n2,
               float* __restrict__ D) {
  const int wave   = threadIdx.x >> 5;
  const int lane   = threadIdx.x & 31;
  const int tile_i = blockIdx.x >> 5;                  // 0..255
  const int tile_j = ((blockIdx.x & 31) << 3) + wave;  // 0..255
  const int itile  = tile_i * 16;
  const int jtile  = tile_j * 16;

  const int m    = lane & 15;
  const int half = lane >> 4;

  const float* arow = f1p + (size_t)(itile + m) * FDIM + 2 * half;
  const float* brow = f2p + (size_t)(jtile + m) * FDIM + 2 * half;

  v8f acc = {0.f, 0.f, 0.f, 0.f, 0.f, 0.f, 0.f, 0.f};
#pragma unroll
  for (int k = 0; k < FDIM; k += 4) {
    v2f a = *(const v2f*)(arow + k);
    v2f b = *(const v2f*)(brow + k);
    acc = __builtin_amdgcn_wmma_f32_16x16x4_f32(
        /*neg_a=*/false, a, /*neg_b=*/false, b,
        /*c_mod=*/(short)0, acc, /*reuse_a=*/false, /*reuse_b=*/false);
  }

  const float n2v = n2[jtile + m];
#pragma unroll
  for (int r = 0; r < 8; ++r) {
    int i = itile + r + 8 * half;
    D[(size_t)i * T_LEN + (jtile + m)] = n1[i] + n2v - 2.0f * acc[r];
  }
}

// ---------------- Phase 4: soft-DTW DP over anti-diagonals ----------------
// Single 1024-thread workgroup (32 wave32 waves). 3 rolling diagonals in LDS (48KB).
// The scattered diagonal gather of D (L2-resident) for diagonal k+1 is issued
// before computing diagonal k, hiding L2 latency behind softmin + barrier.
// R[i,j-1] is carried in a register (same thread owns index i every iteration).
__global__ void __launch_bounds__(1024)
sdtw_dp(const float* __restrict__ D, float* __restrict__ out) {
  __shared__ float buf[3][T_LEN];
  const int tid = threadIdx.x;
  for (int i = tid; i < T_LEN; i += 1024) {
    buf[0][i] = BIGF;   // diag k-2
    buf[1][i] = BIGF;   // diag k-1
  }
  __syncthreads();

  const float* Drow[4];
  float dnext[4], rprev[4];
#pragma unroll
  for (int ib = 0; ib < 4; ++ib) {
    int i = tid + ib * 1024;
    Drow[ib]  = D + (size_t)i * T_LEN;
    rprev[ib] = BIGF;          // r1[i] at k=0
    dnext[ib] = Drow[ib][0];   // k=0: j = -i clamped to 0 (only i==0 is valid anyway)
  }

  int p = 0;  // r2 = buf[p], r1 = buf[(p+1)%3], rk = buf[(p+2)%3]
  for (int k = 0; k < 2 * T_LEN - 1; ++k) {
    float dcur[4];
#pragma unroll
    for (int ib = 0; ib < 4; ++ib) dcur[ib] = dnext[ib];

    // ---- prefetch diagonal k+1 (unconditional, clamped in-bounds) ----
#pragma unroll
    for (int ib = 0; ib < 4; ++ib) {
      int i  = tid + ib * 1024;
      int jn = k + 1 - i;
      jn = jn < 0 ? 0 : (jn > T_LEN - 1 ? T_LEN - 1 : jn);
      dnext[ib] = Drow[ib][jn];
    }

    const float* r2 = buf[p];
    const float* r1 = buf[(p + 1 >= 3) ? p - 2 : p + 1];
    float*       rk = buf[(p + 2 >= 3) ? p - 1 : p + 2];
#pragma unroll
    for (int ib = 0; ib < 4; ++ib) {
      int i = tid + ib * 1024;
      int j = k - i;
      float r2s = (i == 0) ? ((k == 0) ? 0.0f : BIGF) : r2[i - 1];  // R[i-1,j-1]
      float r1s = (i == 0) ? BIGF : r1[i - 1];                      // R[i-1,j]
      float r1c = rprev[ib];                                        // R[i,j-1] (register)
      // -gamma*logsumexp(-z/gamma), gamma=1, max-shifted so BIG underflows to 0
      float mn = fminf(r2s, fminf(r1s, r1c));
      float s  = __expf(mn - r2s) + __expf(mn - r1s) + __expf(mn - r1c);
      float rkv = (j >= 0 && j < T_LEN) ? (dcur[ib] + mn - __logf(s)) : BIGF;
      rk[i]     = rkv;
      rprev[ib] = rkv;
    }
    __syncthreads();
    p = (p == 2) ? 0 : p + 1;
  }

  // R[4095,4095] was computed at k=8190 by thread 1023, ib=3 (i=4095)
  if (tid == 1023) out[0] = rprev[3];
}

// ---------------- launch ----------------
extern "C" void kernel_launch(void* const* d_in, const int* in_sizes, int n_in,
                              void* d_out, int out_size, void* d_ws, size_t ws_size,
                              hipStream_t stream) {
  (void)in_sizes; (void)n_in; (void)out_size; (void)ws_size;
  const float* x = (const float*)d_in[0];
  const float* y = (const float*)d_in[1];

  float* ws  = (float*)d_ws;
  float* f1p = ws;                         // 4096*72
  float* f2p = f1p + T_LEN * FDIM;         // 4096*72
  float* n1  = f2p + T_LEN * FDIM;         // 4096
  float* n2  = n1 + T_LEN;                 // 4096
  float* D   = n2 + T_LEN;                 // 4096*4096 (~64MB; total ws ~66.3MB)

  {
    int total = 2 * T_LEN * FDIM;
    sdtw_prep<<<(total + 255) / 256, 256, 0, stream>>>(x, y, f1p, f2p);
  }
  sdtw_norms<<<(2 * T_LEN) / 8, 256, 0, stream>>>(f1p, f2p, n1, n2);
  sdtw_dist_wmma<<<256 * 32, 256, 0, stream>>>(f1p, f2p, n1, n2, D);
  sdtw_dp<<<1, 1024, 0, stream>>>(D, (float*)d_out);
}